// MultiHeadAttention_encoder_88656714924834
// MI455X (gfx1250) — compile-verified
//
#include <hip/hip_runtime.h>
#include <hip/hip_bf16.h>
#include <math.h>

typedef __attribute__((ext_vector_type(2))) float v2f;
typedef __attribute__((ext_vector_type(8))) float v8f;

#define D_MODEL 64
#define DK      16
#define NH      4
#define NB      32
#define SS      1024
#define BH      (NB * NH)   /* 128 */
#define QT      (SS / 16)   /* 64  */

__device__ __forceinline__ v8f wmma4(v2f a, v2f b, v8f c) {
  // D = A(16x4 f32) * B(4x16 f32) + C(16x16 f32)
  return __builtin_amdgcn_wmma_f32_16x16x4_f32(false, a, false, b, (short)0, c,
                                               false, false);
}

// ---------------------------------------------------------------------------
// Kernel 1: QKV projection.  Out[(b*4+h)][s][d] = sum_c X[b][s][c]*W[c][h*16+d]
// ---------------------------------------------------------------------------
__global__ __launch_bounds__(256) void proj_kernel(const float* __restrict__ X,
                                                   const float* __restrict__ W,
                                                   float* __restrict__ Out) {
  int idx = blockIdx.x * 256 + threadIdx.x;  // 0 .. B*S*64
  int j   = idx & 63;
  int bs  = idx >> 6;           // b*1024 + s
  int b   = bs >> 10;
  int s   = bs & 1023;
  int h   = j >> 4;
  int d   = j & 15;
  const float* xrow = X + (size_t)bs * 64;
  float acc = 0.f;
#pragma unroll
  for (int c = 0; c < 64; ++c) acc = fmaf(xrow[c], W[c * 64 + j], acc);
  Out[(((size_t)(b * NH + h)) * SS + s) * DK + d] = acc;
}

// ---------------------------------------------------------------------------
// Kernel 2: fused attention.  One workgroup = (bh, 16-row q tile), 4 waves
// split the key range.  Two passes over bias: stats, then emit attn + P*V.
// Pass 1: per-lane online (m,l), single exp per element; 16-lane merge once.
// Mask predicates cached in a 128-bit register *shift register* (statically
// indexed -> no cndmask trees, loops stay rolled -> low VGPR pressure).
// Pass 2: p = exp(val - (m* + ln l*)) -- normalization folded into the shift.
// ---------------------------------------------------------------------------
__global__ __launch_bounds__(128) void attn_kernel(
    const float* __restrict__ Qp, const float* __restrict__ Kp,
    const float* __restrict__ Vp, const float* __restrict__ bias,
    const unsigned char* __restrict__ mask, float* __restrict__ attn_out,
    float* __restrict__ ctx_out) {
  __shared__ float sm[4][16];
  __shared__ float sl[4][16];
  __shared__ float pt[4][16][20];   // per-wave P tile, pad 20 -> no bank conflicts
  __shared__ float cacc[16][20];    // cross-wave context accumulator

  const int qt = blockIdx.x;
  const int bh = blockIdx.y;
  const int b  = bh >> 2;
  const int h  = bh & 3;
  const int tid  = threadIdx.x;
  const int wave = tid >> 5;
  const int lane = tid & 31;
  const int half = lane >> 4;
  const int l15  = lane & 15;

  for (int i = tid; i < 16 * 20; i += 128) ((float*)cacc)[i] = 0.f;

  // Q tile in WMMA A layout (M=l15, K = 4s + 2*half + j), kept in regs.
  float aq[8];
  {
    const float* qbase = Qp + ((size_t)bh * SS + qt * 16 + l15) * DK;
#pragma unroll
    for (int s = 0; s < 4; ++s) {
      aq[2 * s + 0] = qbase[4 * s + 2 * half + 0];
      aq[2 * s + 1] = qbase[4 * s + 2 * half + 1];
    }
  }

  const int kt0 = wave * 16;  // this wave's 16 k-tiles (256 keys)
  float mrun[8], lrun[8];
#pragma unroll
  for (int v = 0; v < 8; ++v) { mrun[v] = -INFINITY; lrun[v] = 0.f; }
  // 128-bit mask shift register: tile t's byte ends up FIFO-ordered.
  unsigned int mq0 = 0u, mq1 = 0u, mq2 = 0u, mq3 = 0u;

  const size_t biasRowBase = ((size_t)bh * SS + (size_t)qt * 16) * SS;
  const size_t maskRowBase = ((size_t)b  * SS + (size_t)qt * 16) * SS;

  // ==== PASS 1: per-lane online row stats (1 exp/elem, no shuffles) ======
  for (int t = 0; t < 16; ++t) {
    const int kt = kt0 + t;
    const int col = kt * 16 + l15;
    __builtin_prefetch(&bias[biasRowBase + (size_t)(8 * half) * SS + col + 16], 0, 1);
    float kb[8];
    const float* kbase = Kp + ((size_t)bh * SS + kt * 16 + l15) * DK;
#pragma unroll
    for (int s = 0; s < 4; ++s) {
      kb[2 * s + 0] = kbase[4 * s + 2 * half + 0];
      kb[2 * s + 1] = kbase[4 * s + 2 * half + 1];
    }
    v8f sc = {};
#pragma unroll
    for (int s = 0; s < 4; ++s) {
      v2f a;  a.x = aq[2 * s];  a.y = aq[2 * s + 1];
      v2f bb; bb.x = kb[2 * s]; bb.y = kb[2 * s + 1];
      sc = wmma4(a, bb, sc);
    }
    unsigned int mbyte = 0u;
#pragma unroll
    for (int v = 0; v < 8; ++v) {
      const int row = v + 8 * half;
      float val = fmaf(sc[v], 0.25f, bias[biasRowBase + (size_t)row * SS + col]);
      unsigned int msk = mask[maskRowBase + (size_t)row * SS + col] ? 1u : 0u;
      mbyte |= msk << v;                          // constant shift
      if (msk) val = -1e9f;
      // single-exp online update: one of the two rescale exps is exp(0)=1
      float diff = val - mrun[v];                 // >0 iff val is the new max
      float e = __expf(-fabsf(diff));
      if (diff > 0.f) {
        lrun[v] = fmaf(lrun[v], e, 1.f);
        mrun[v] = val;
      } else {
        lrun[v] = lrun[v] + e;
      }
    }
    // shift the 128-bit mask register left by 8 and insert this tile's byte
    mq3 = (mq3 << 8) | (mq2 >> 24);
    mq2 = (mq2 << 8) | (mq1 >> 24);
    mq1 = (mq1 << 8) | (mq0 >> 24);
    mq0 = (mq0 << 8) | mbyte;
  }

  // cross-column (16-lane) merge of per-lane (m,l), once per wave
#pragma unroll
  for (int off = 1; off < 16; off <<= 1) {
#pragma unroll
    for (int v = 0; v < 8; ++v) {
      float om = __shfl_xor(mrun[v], off, 32);
      float ol = __shfl_xor(lrun[v], off, 32);
      float mn = fmaxf(mrun[v], om);
      lrun[v] = lrun[v] * __expf(mrun[v] - mn) + ol * __expf(om - mn);
      mrun[v] = mn;
    }
  }

  // cross-wave (m,l) merge; fold normalization into a per-row shift
  if (l15 == 0) {
#pragma unroll
    for (int v = 0; v < 8; ++v) {
      sm[wave][v + 8 * half] = mrun[v];
      sl[wave][v + 8 * half] = lrun[v];
    }
  }
  __syncthreads();
  float shift[8];
#pragma unroll
  for (int v = 0; v < 8; ++v) {
    const int row = v + 8 * half;
    float mstar = fmaxf(fmaxf(sm[0][row], sm[1][row]), fmaxf(sm[2][row], sm[3][row]));
    float lstar = 0.f;
#pragma unroll
    for (int w = 0; w < 4; ++w) lstar += sl[w][row] * __expf(sm[w][row] - mstar);
    shift[v] = mstar + __logf(lstar);   // p = exp(val - shift)
  }

  // ================= PASS 2: emit attn, accumulate P*V ===================
  v8f ctx = {};
  for (int t = 0; t < 16; ++t) {
    const int kt = kt0 + t;
    const int col = kt * 16 + l15;
    __builtin_prefetch(&bias[biasRowBase + (size_t)(8 * half) * SS + col + 16], 0, 1);
    float kb[8];
    const float* kbase = Kp + ((size_t)bh * SS + kt * 16 + l15) * DK;
#pragma unroll
    for (int s = 0; s < 4; ++s) {
      kb[2 * s + 0] = kbase[4 * s + 2 * half + 0];
      kb[2 * s + 1] = kbase[4 * s + 2 * half + 1];
    }
    v8f sc = {};
#pragma unroll
    for (int s = 0; s < 4; ++s) {
      v2f a;  a.x = aq[2 * s];  a.y = aq[2 * s + 1];
      v2f bb; bb.x = kb[2 * s]; bb.y = kb[2 * s + 1];
      sc = wmma4(a, bb, sc);
    }
    // peel tile t's mask byte (FIFO) and advance the shift register
    unsigned int mbyte = mq3 >> 24;
    mq3 = (mq3 << 8) | (mq2 >> 24);
    mq2 = (mq2 << 8) | (mq1 >> 24);
    mq1 = (mq1 << 8) | (mq0 >> 24);
    mq0 = (mq0 << 8);
#pragma unroll
    for (int v = 0; v < 8; ++v) {
      const int row = v + 8 * half;
      float val = fmaf(sc[v], 0.25f, bias[biasRowBase + (size_t)row * SS + col]);
      if ((mbyte >> v) & 1u) val = -1e9f;         // constant shift
      float p = __expf(val - shift[v]);
      attn_out[biasRowBase + (size_t)row * SS + col] = p;
      pt[wave][row][l15] = p;   // stash in C layout
    }
    // read back in A layout (DS ops in-order within a wave)
    float pa[8];
#pragma unroll
    for (int s = 0; s < 4; ++s) {
      pa[2 * s + 0] = pt[wave][l15][4 * s + 2 * half + 0];
      pa[2 * s + 1] = pt[wave][l15][4 * s + 2 * half + 1];
    }
    const float* vbase = Vp + ((size_t)bh * SS + kt * 16) * DK;
#pragma unroll
    for (int s = 0; s < 4; ++s) {
      v2f a;  a.x = pa[2 * s]; a.y = pa[2 * s + 1];
      v2f bb;
      bb.x = vbase[(4 * s + 2 * half + 0) * DK + l15];
      bb.y = vbase[(4 * s + 2 * half + 1) * DK + l15];
      ctx = wmma4(a, bb, ctx);
    }
  }

  // cross-wave context reduce, then write [16 x 16] head context to ws
#pragma unroll
  for (int v = 0; v < 8; ++v) atomicAdd(&cacc[v + 8 * half][l15], ctx[v]);
  __syncthreads();
  for (int i = tid; i < 256; i += 128) {
    int row = i >> 4, colc = i & 15;
    ctx_out[((size_t)b * SS + qt * 16 + row) * D_MODEL + h * 16 + colc] =
        cacc[row][colc];
  }
}

// ---------------------------------------------------------------------------
// Kernel 3: out = LN(ctx @ W_fc + input_Q).  Wave w computes cols [16w,16w+16).
// ---------------------------------------------------------------------------
__global__ __launch_bounds__(128) void out_kernel(
    const float* __restrict__ ctx, const float* __restrict__ Wfc,
    const float* __restrict__ inQ, const float* __restrict__ gamma,
    const float* __restrict__ beta, float* __restrict__ out) {
  __shared__ float xt[16][68];
  const int st = blockIdx.x;   // s tile
  const int b  = blockIdx.y;
  const int tid  = threadIdx.x;
  const int wave = tid >> 5, lane = tid & 31, half = lane >> 4, l15 = lane & 15;

  const float* arow = ctx + ((size_t)b * SS + st * 16 + l15) * D_MODEL;
  v8f acc = {};
#pragma unroll
  for (int s = 0; s < 16; ++s) {
    v2f a;  a.x = arow[4 * s + 2 * half]; a.y = arow[4 * s + 2 * half + 1];
    v2f bb;
    bb.x = Wfc[(4 * s + 2 * half + 0) * 64 + wave * 16 + l15];
    bb.y = Wfc[(4 * s + 2 * half + 1) * 64 + wave * 16 + l15];
    acc = wmma4(a, bb, acc);
  }
#pragma unroll
  for (int v = 0; v < 8; ++v) {
    const int row = v + 8 * half;
    float x = acc[v] +
              inQ[((size_t)b * SS + st * 16 + row) * 64 + wave * 16 + l15];
    xt[row][wave * 16 + l15] = x;
  }
  __syncthreads();

  const int r = tid >> 3, g = tid & 7;
  float s1 = 0.f, s2 = 0.f, xv[8];
#pragma unroll
  for (int j = 0; j < 8; ++j) {
    xv[j] = xt[r][g * 8 + j];
    s1 += xv[j];
    s2 += xv[j] * xv[j];
  }
#pragma unroll
  for (int off = 1; off < 8; off <<= 1) {
    s1 += __shfl_xor(s1, off, 32);
    s2 += __shfl_xor(s2, off, 32);
  }
  float mu   = s1 * (1.f / 64.f);
  float var  = s2 * (1.f / 64.f) - mu * mu;
  float rstd = rsqrtf(var + 1e-5f);
  float* orow = out + ((size_t)b * SS + st * 16 + r) * 64;
#pragma unroll
  for (int j = 0; j < 8; ++j) {
    int c = g * 8 + j;
    orow[c] = (xv[j] - mu) * rstd * gamma[c] + beta[c];
  }
}

// ---------------------------------------------------------------------------
extern "C" void kernel_launch(void* const* d_in, const int* in_sizes, int n_in,
                              void* d_out, int out_size, void* d_ws,
                              size_t ws_size, hipStream_t stream) {
  (void)in_sizes; (void)n_in; (void)out_size; (void)ws_size;
  const float* inQ  = (const float*)d_in[0];
  const float* inK  = (const float*)d_in[1];
  const float* inV  = (const float*)d_in[2];
  const unsigned char* mask = (const unsigned char*)d_in[3];
  const float* bias = (const float*)d_in[4];
  const float* WQ   = (const float*)d_in[5];
  const float* WK   = (const float*)d_in[6];
  const float* WV   = (const float*)d_in[7];
  const float* Wfc  = (const float*)d_in[8];
  const float* gam  = (const float*)d_in[9];
  const float* bet  = (const float*)d_in[10];

  float* out  = (float*)d_out;                                 // [32,1024,64]
  float* attn = (float*)d_out + (size_t)NB * SS * D_MODEL;     // [32,4,1024,1024]

  float* ws  = (float*)d_ws;
  float* Qp  = ws;                                             // [128][1024][16]
  float* Kp  = Qp + (size_t)BH * SS * DK;
  float* Vp  = Kp + (size_t)BH * SS * DK;
  float* ctx = Vp + (size_t)BH * SS * DK;                      // [32][1024][64]

  dim3 pg((NB * SS * D_MODEL) / 256);
  proj_kernel<<<pg, dim3(256), 0, stream>>>(inQ, WQ, Qp);
  proj_kernel<<<pg, dim3(256), 0, stream>>>(inK, WK, Kp);
  proj_kernel<<<pg, dim3(256), 0, stream>>>(inV, WV, Vp);

  attn_kernel<<<dim3(QT, BH), dim3(128), 0, stream>>>(Qp, Kp, Vp, bias, mask,
                                                      attn, ctx);

  out_kernel<<<dim3(QT, NB), dim3(128), 0, stream>>>(ctx, Wfc, inQ, gam, bet,
                                                     out);
}